// Loss_18047452578593
// MI455X (gfx1250) — compile-verified
//
#include <hip/hip_runtime.h>
#include <stdint.h>

#define T_DIM 4096
#define V_DIM 32000
#define S_DIM 256
#define L_DIM 513              // 2*S+1
#define L_PAD 516              // multiple of 4 -> rows are 2064B = 129 * 16B
#define NEGF  (-1.0e30f)
#define LOG2E 1.4426950408889634f
#define LN2   0.6931471805599453f

typedef __attribute__((ext_vector_type(2))) float v2f;
typedef __attribute__((ext_vector_type(4))) float v4f;
typedef __attribute__((ext_vector_type(8))) float v8f;

// ---------------------------------------------------------------------------
// Kernel 0: build extended label sequence and skip flags (tiny).
// ext[l] = (l odd) ? target[min(l/2,S-1)] : 0 ; skip = (ext!=0 && ext!=ext[l-2])
// ---------------------------------------------------------------------------
__global__ __launch_bounds__(512)
void ctc_prep(const int* __restrict__ target, int* __restrict__ ext,
              int* __restrict__ skip) {
  for (int l = threadIdx.x; l < L_PAD; l += 512) {
    int e = 0, sk = 0;
    if (l < L_DIM) {
      int i0 = l >> 1; if (i0 > S_DIM - 1) i0 = S_DIM - 1;
      e = (l & 1) ? target[i0] : 0;
      int e2 = -1;                       // pad value for l-2 < 0
      if (l >= 2) {
        int lm = l - 2;
        int i2 = lm >> 1; if (i2 > S_DIM - 1) i2 = S_DIM - 1;
        e2 = (lm & 1) ? target[i2] : 0;
      }
      sk = (e != 0 && e != e2) ? 1 : 0;
    }
    ext[l] = e; skip[l] = sk;
  }
}

// ---------------------------------------------------------------------------
// Kernel 1: per-timestep streaming logsumexp (log2 domain) + label gather.
// One block per t, 256 threads. NT b128 stream, online lse, WMMA f32 tail
// reduction, writes lp_ext[t][l] = row[ext[l]]*log2e - lse2.
// ---------------------------------------------------------------------------
__global__ __launch_bounds__(256)
void ctc_row_lse(const float* __restrict__ lp, const int* __restrict__ ext,
                 float* __restrict__ lp_ext) {
  __shared__ float gbuf[L_PAD];
  __shared__ float red[256];
  __shared__ float wred[8];
  __shared__ float lse_sh;

  const int tid  = threadIdx.x;
  const int lane = tid & 31;
  const int wid  = tid >> 5;
  const int t    = blockIdx.x;
  const float* __restrict__ row = lp + (size_t)t * V_DIM;

  // gather label logits first (RT hint -> lines parked in L2 for the stream)
  for (int l = tid; l < L_DIM; l += 256) gbuf[l] = row[ext[l]];

  // streaming online logsumexp, base-2 domain, non-temporal b128 loads
  const v4f* row4 = (const v4f*)row;
  float m = NEGF, s = 0.0f;
  for (int i = tid; i < V_DIM / 4; i += 256) {
    v4f x = __builtin_nontemporal_load(&row4[i]);
    float y0 = x.x * LOG2E, y1 = x.y * LOG2E;
    float y2 = x.z * LOG2E, y3 = x.w * LOG2E;
    float mx = fmaxf(fmaxf(y0, y1), fmaxf(y2, y3));
    if (mx > m) { s *= __builtin_amdgcn_exp2f(m - mx); m = mx; }  // rare
    s += __builtin_amdgcn_exp2f(y0 - m);
    s += __builtin_amdgcn_exp2f(y1 - m);
    s += __builtin_amdgcn_exp2f(y2 - m);
    s += __builtin_amdgcn_exp2f(y3 - m);
  }

  // block max: wave32 shuffle tree, then 8 wave maxima
  float wm = m;
  for (int off = 16; off > 0; off >>= 1)
    wm = fmaxf(wm, __shfl_xor(wm, off, 32));
  if (lane == 0) wred[wid] = wm;
  __syncthreads();
  float M = fmaxf(fmaxf(fmaxf(wred[0], wred[1]), fmaxf(wred[2], wred[3])),
                  fmaxf(fmaxf(wred[4], wred[5]), fmaxf(wred[6], wred[7])));

  red[tid] = s * __builtin_amdgcn_exp2f(m - M);
  __syncthreads();
  if (tid < 64)
    red[tid] = red[tid] + red[tid + 64] + red[tid + 128] + red[tid + 192];
  __syncthreads();

  // 64 fp32 partials -> one V_WMMA_F32_16X16X4_F32 with B = ones:
  // D[m][n] = sum_k A[m][k]; C layout: lane<16 VGPR j = row j, lane>=16 = row j+8
  if (tid < 32) {                       // whole wave 0 -> EXEC all ones
    v2f a, b; v8f c = {};
    b.x = 1.0f; b.y = 1.0f;
    int r = lane & 15;
    int kb = (lane < 16) ? 0 : 2;       // A layout: lanes16-31 hold K=2,3
    a.x = red[r * 4 + kb];
    a.y = red[r * 4 + kb + 1];
    c = __builtin_amdgcn_wmma_f32_16x16x4_f32(false, a, false, b,
                                              (short)0, c, false, false);
    float t8 = ((c[0] + c[1]) + (c[2] + c[3])) + ((c[4] + c[5]) + (c[6] + c[7]));
    if (lane == 0)  wred[0] = t8;       // rows 0..7
    if (lane == 16) wred[1] = t8;       // rows 8..15
  }
  __syncthreads();
  if (tid == 0)
    lse_sh = M + __builtin_amdgcn_logf(wred[0] + wred[1]);  // v_log_f32 = log2
  __syncthreads();

  const float lse2 = lse_sh;
  float* __restrict__ dst = lp_ext + (size_t)t * L_PAD;
  for (int l = tid; l < L_PAD; l += 256)
    dst[l] = (l < L_DIM) ? (gbuf[l] * LOG2E - lse2) : NEGF;
}

// ---------------------------------------------------------------------------
// Kernel 2: serial alpha recursion. One workgroup (16 waves / one WGP).
// Alpha double-buffered in LDS; next lp_ext row prefetched with
// global_load_async_to_lds_b128 and synced with s_wait_asynccnt.
// ---------------------------------------------------------------------------
__global__ __launch_bounds__(512)
void ctc_alpha(const float* __restrict__ lp_ext, const int* __restrict__ skip,
               float* __restrict__ out) {
  __shared__ float alpha[2][L_DIM + 3];   // [2+l]; slots 0,1 = NEG padding
  __shared__ float lpb[2][L_PAD];         // double-buffered lp_ext rows

  const int tid = threadIdx.x;
  const int l0  = tid;                    // every thread owns l0
  const int l1  = tid + 512;              // only tid==0 also owns l=512
  const int sk0 = skip[l0];
  const int sk1 = (l1 < L_DIM) ? skip[l1] : 0;
  const int NPF = (L_PAD * 4) / 16;       // 129 b128 transfers per row

  // init alpha(t=0) = NEG everywhere; load row 0 sync, prefetch row 1 async
  for (int l = tid; l < L_DIM + 3; l += 512) alpha[0][l] = NEGF;
  for (int l = tid; l < L_PAD; l += 512)     lpb[0][l] = lp_ext[l];
  if (tid < NPF) {
    unsigned lds_off = (unsigned)(uintptr_t)&lpb[1][tid * 4];
    uint64_t ga = (uint64_t)(lp_ext + L_PAD + tid * 4);
    asm volatile("global_load_async_to_lds_b128 %0, %1, off"
                 :: "v"(lds_off), "v"(ga) : "memory");
  }
  __syncthreads();
  if (tid == 0) {
    alpha[0][2 + 0] = lpb[0][0];
    alpha[0][2 + 1] = lpb[0][1];
  }
  asm volatile("s_wait_asynccnt 0" ::: "memory");
  __syncthreads();

  int p = 0;
  for (int t = 1; t < T_DIM; ++t) {
    const int cur = t & 1, nxt = cur ^ 1;
    // prefetch row t+1 into the buffer not being read this step
    if (t + 1 < T_DIM && tid < NPF) {
      unsigned lds_off = (unsigned)(uintptr_t)&lpb[nxt][tid * 4];
      uint64_t ga = (uint64_t)(lp_ext + (size_t)(t + 1) * L_PAD + tid * 4);
      asm volatile("global_load_async_to_lds_b128 %0, %1, off"
                   :: "v"(lds_off), "v"(ga) : "memory");
    }
    const float* __restrict__ A  = alpha[p];
    float* __restrict__       Bv = alpha[p ^ 1];
    {
      float a = A[2 + l0], b = A[1 + l0];
      float c = sk0 ? A[l0] : NEGF;
      float mm = fmaxf(a, fmaxf(b, c));
      float ss = __builtin_amdgcn_exp2f(a - mm) +
                 __builtin_amdgcn_exp2f(b - mm) +
                 __builtin_amdgcn_exp2f(c - mm);
      Bv[2 + l0] = mm + __builtin_amdgcn_logf(ss) + lpb[cur][l0];
    }
    if (l1 < L_DIM) {
      float a = A[2 + l1], b = A[1 + l1];
      float c = sk1 ? A[l1] : NEGF;
      float mm = fmaxf(a, fmaxf(b, c));
      float ss = __builtin_amdgcn_exp2f(a - mm) +
                 __builtin_amdgcn_exp2f(b - mm) +
                 __builtin_amdgcn_exp2f(c - mm);
      Bv[2 + l1] = mm + __builtin_amdgcn_logf(ss) + lpb[cur][l1];
    }
    if (tid == 0) { Bv[0] = NEGF; Bv[1] = NEGF; }
    asm volatile("s_wait_asynccnt 0" ::: "memory");
    __syncthreads();
    p ^= 1;
  }

  if (tid == 0) {
    float a  = alpha[p][2 + L_DIM - 1];
    float b  = alpha[p][2 + L_DIM - 2];
    float mm = fmaxf(a, b);
    float ll2 = mm + __builtin_amdgcn_logf(__builtin_amdgcn_exp2f(a - mm) +
                                           __builtin_amdgcn_exp2f(b - mm));
    out[0] = -(ll2 * LN2) / (float)S_DIM;   // mean over B==1
  }
}

// ---------------------------------------------------------------------------
extern "C" void kernel_launch(void* const* d_in, const int* in_sizes, int n_in,
                              void* d_out, int out_size, void* d_ws, size_t ws_size,
                              hipStream_t stream) {
  const float* log_probs = (const float*)d_in[0];   // (T,B,V) f32
  const int*   target    = (const int*)d_in[1];     // (B,S) i32
  float*       out       = (float*)d_out;           // scalar f32

  // workspace: ext[516] int | skip[516] int | lp_ext[T*516] f32 (~8.5 MB)
  char* ws = (char*)d_ws;
  int*   ext    = (int*)ws;
  int*   skip   = (int*)(ws + (size_t)L_PAD * 4);
  float* lp_ext = (float*)(ws + (size_t)2 * L_PAD * 4);

  ctc_prep   <<<1,      512, 0, stream>>>(target, ext, skip);
  ctc_row_lse<<<T_DIM,  256, 0, stream>>>(log_probs, ext, lp_ext);
  ctc_alpha  <<<1,      512, 0, stream>>>(lp_ext, skip, out);
}